// CBAM3D_16484084483200
// MI455X (gfx1250) — compile-verified
//
#include <hip/hip_runtime.h>
#include <hip/hip_bf16.h>
#include <math.h>

// ---------------------------------------------------------------------------
// CBAM3D: x[2,64,64,64,64] fp32.
// Bandwidth-bound (~660MB traffic -> ~28us @ 23.3TB/s). Channel MLP mapped to
// V_WMMA_F32_16X16X4_F32 (the only real matmul in the op).
// ---------------------------------------------------------------------------

#define NB 2
#define NC 64
#define SPAT (64*64*64)          // 262144 spatial elems per (b,c)
#define SPAT4 (SPAT/4)           // 65536 float4
#define SPLIT 8                  // partial-reduction splits per (b,c)

typedef float v2f __attribute__((ext_vector_type(2)));
typedef float v8f __attribute__((ext_vector_type(8)));

__device__ __forceinline__ float sigmoidf_(float v) { return 1.0f / (1.0f + expf(-v)); }

// ---------------------------------------------------------------------------
// K1: per-(b,c) partial sum & max over a 1/8 chunk of the 64^3 volume.
// grid = 128*SPLIT blocks, 256 threads, float4 streaming loads.
// ---------------------------------------------------------------------------
__global__ void __launch_bounds__(256) cbam_stats(const float4* __restrict__ x4,
                                                  float* __restrict__ psum,
                                                  float* __restrict__ pmax) {
    const int t = threadIdx.x;
    const int bc = blockIdx.x >> 3;          // 0..127
    const int p  = blockIdx.x & (SPLIT - 1); // 0..7
    const int base = bc * SPAT4 + p * (SPAT4 / SPLIT);

    float s = 0.0f;
    float m = -3.402823466e38f;
    #pragma unroll 4
    for (int i = t; i < SPAT4 / SPLIT; i += 256) {
        float4 v = x4[base + i];
        s += (v.x + v.y) + (v.z + v.w);
        m = fmaxf(m, fmaxf(fmaxf(v.x, v.y), fmaxf(v.z, v.w)));
    }

    __shared__ float rs[256];
    __shared__ float rm[256];
    rs[t] = s; rm[t] = m;
    __syncthreads();
    for (int off = 128; off > 0; off >>= 1) {
        if (t < off) { rs[t] += rs[t + off]; rm[t] = fmaxf(rm[t], rm[t + off]); }
        __syncthreads();
    }
    if (t == 0) { psum[blockIdx.x] = rs[0]; pmax[blockIdx.x] = rm[0]; }
}

// ---------------------------------------------------------------------------
// K2: single wave32. Combine partials -> stats matrix V[64][4]
//    (cols: avg_b0, avg_b1, max_b0, max_b1), then
//    fc1 (8x64 * 64x4) and fc2 (64x8 * 8x4) via V_WMMA_F32_16X16X4_F32,
//    ReLU between, sigmoid(avg_path + max_path) -> ca[2][64].
// A 16x4 layout: lane m=l&15, hi=l>>4: a = {A[m][2*hi], A[m][2*hi+1]}
// B 4x16 layout: n=l&15:               b = {B[2*hi][n], B[2*hi+1][n]}
// C/D 16x16:     elem e: (M=e+8*hi, N=l&15)
// ---------------------------------------------------------------------------
__global__ void __launch_bounds__(32) cbam_mlp_wmma(const float* __restrict__ psum,
                                                    const float* __restrict__ pmax,
                                                    const float* __restrict__ w1,  // [8][64]
                                                    const float* __restrict__ w2,  // [64][8]
                                                    float* __restrict__ ca) {      // [2][64]
    __shared__ float V[NC * 4];   // [c][j]
    __shared__ float H[8 * 16];   // relu(fc1) as [r][n] (B-source for fc2)
    __shared__ float O[NC * 4];   // fc2 out [c][j]

    const int lane = threadIdx.x;

    // combine partials: 128 (b,c) entries, 4 per lane
    for (int g = 0; g < 4; ++g) {
        int bc = g * 32 + lane;
        float s = 0.0f, m = -3.402823466e38f;
        #pragma unroll
        for (int p = 0; p < SPLIT; ++p) {
            s += psum[bc * SPLIT + p];
            m = fmaxf(m, pmax[bc * SPLIT + p]);
        }
        int b = bc >> 6, c = bc & 63;
        V[c * 4 + b]     = s * (1.0f / (float)SPAT);  // avg
        V[c * 4 + 2 + b] = m;                         // max
    }
    __syncthreads();

    const int m  = lane & 15;
    const int hi = lane >> 4;

    // ---- fc1: out1[16pad x 16pad] = w1[8x64] * V[64x4pad], K=64 in 16 steps
    v8f acc = {0.f, 0.f, 0.f, 0.f, 0.f, 0.f, 0.f, 0.f};
    for (int kb = 0; kb < 64; kb += 4) {
        int k0 = kb + 2 * hi;
        float a0 = w1[min(m, 7) * 64 + k0];
        float a1 = w1[min(m, 7) * 64 + k0 + 1];
        float b0 = V[k0 * 4 + min(m, 3)];
        float b1 = V[(k0 + 1) * 4 + min(m, 3)];
        v2f a, b;
        a.x = (m < 8) ? a0 : 0.f;  a.y = (m < 8) ? a1 : 0.f;
        b.x = (m < 4) ? b0 : 0.f;  b.y = (m < 4) ? b1 : 0.f;
        acc = __builtin_amdgcn_wmma_f32_16x16x4_f32(false, a, false, b,
                                                    (short)0, acc, false, false);
    }
    // ReLU; rows 0..7 live in lanes 0-15 (M=e), re-layout to LDS for fc2's B
    if (hi == 0) {
        #pragma unroll
        for (int e = 0; e < 8; ++e) H[e * 16 + m] = fmaxf(acc[e], 0.0f);
    }
    __syncthreads();

    // ---- fc2: out2[64 x 4] = w2[64x8] * H[8x4pad]; 4 M-tiles, K=8 in 2 steps
    for (int ct = 0; ct < 4; ++ct) {
        v8f a2 = {0.f, 0.f, 0.f, 0.f, 0.f, 0.f, 0.f, 0.f};
        #pragma unroll
        for (int k0 = 0; k0 < 8; k0 += 4) {
            int kk = k0 + 2 * hi;
            v2f a, b;
            a.x = w2[(ct * 16 + m) * 8 + kk];
            a.y = w2[(ct * 16 + m) * 8 + kk + 1];
            b.x = H[kk * 16 + m];
            b.y = H[(kk + 1) * 16 + m];
            a2 = __builtin_amdgcn_wmma_f32_16x16x4_f32(false, a, false, b,
                                                       (short)0, a2, false, false);
        }
        if (m < 4) {
            #pragma unroll
            for (int e = 0; e < 8; ++e) O[(ct * 16 + e + 8 * hi) * 4 + m] = a2[e];
        }
    }
    __syncthreads();

    // ca[b][c] = sigmoid(mlp(avg) + mlp(max))
    for (int g = 0; g < 4; ++g) {
        int idx = g * 32 + lane;           // = b*64 + c
        int b = idx >> 6, c = idx & 63;
        float v = O[c * 4 + b] + O[c * 4 + 2 + b];
        ca[idx] = sigmoidf_(v);
    }
}

// ---------------------------------------------------------------------------
// K3: spatial stats of channel-gated x. grid(256, 2), 256 thr, float4/thread.
// avg_s/max_s over the 64 channels of x[b][c][s] * ca[b][c].
// ---------------------------------------------------------------------------
__global__ void __launch_bounds__(256) cbam_spatial_stats(const float4* __restrict__ x4,
                                                          const float* __restrict__ ca,
                                                          float4* __restrict__ avg_s4,
                                                          float4* __restrict__ max_s4) {
    const int t = threadIdx.x;
    const int b = blockIdx.y;
    const int s4 = blockIdx.x * 256 + t;   // 0..65535

    __shared__ float ca_lds[NC];
    if (t < NC) ca_lds[t] = ca[b * NC + t];
    __syncthreads();

    float4 sum = make_float4(0.f, 0.f, 0.f, 0.f);
    float4 mx  = make_float4(-3.402823466e38f, -3.402823466e38f,
                             -3.402823466e38f, -3.402823466e38f);
    #pragma unroll 4
    for (int c = 0; c < NC; ++c) {
        float g = ca_lds[c];
        float4 v = x4[(b * NC + c) * SPAT4 + s4];
        float vx = v.x * g, vy = v.y * g, vz = v.z * g, vw = v.w * g;
        sum.x += vx; sum.y += vy; sum.z += vz; sum.w += vw;
        mx.x = fmaxf(mx.x, vx); mx.y = fmaxf(mx.y, vy);
        mx.z = fmaxf(mx.z, vz); mx.w = fmaxf(mx.w, vw);
    }
    const float inv = 1.0f / (float)NC;
    avg_s4[b * SPAT4 + s4] = make_float4(sum.x * inv, sum.y * inv, sum.z * inv, sum.w * inv);
    max_s4[b * SPAT4 + s4] = mx;
}

// ---------------------------------------------------------------------------
// K4: 7x7x7 2-channel correlation + sigmoid. 8^3 output tile, 14^3 halo in LDS.
// grid(512, 2), 256 threads, 2 outputs/thread.
// ---------------------------------------------------------------------------
__global__ void __launch_bounds__(256) cbam_conv7(const float* __restrict__ avg_s,
                                                  const float* __restrict__ max_s,
                                                  const float* __restrict__ wsp, // [2][7][7][7]
                                                  float* __restrict__ sig_sa) {
    __shared__ float sA[14 * 14 * 14];
    __shared__ float sM[14 * 14 * 14];
    __shared__ float wl[2 * 343];

    const int t = threadIdx.x;
    const int b = blockIdx.y;
    const int tz = blockIdx.x >> 6;
    const int ty = (blockIdx.x >> 3) & 7;
    const int tx = blockIdx.x & 7;

    for (int i = t; i < 686; i += 256) wl[i] = wsp[i];

    for (int idx = t; idx < 14 * 14 * 14; idx += 256) {
        int lz = idx / 196, r = idx % 196, ly = r / 14, lx = r % 14;
        int gz = tz * 8 - 3 + lz, gy = ty * 8 - 3 + ly, gx = tx * 8 - 3 + lx;
        bool ok = ((unsigned)gz < 64u) & ((unsigned)gy < 64u) & ((unsigned)gx < 64u);
        int off = b * SPAT + (gz * 64 + gy) * 64 + gx;
        sA[idx] = ok ? avg_s[off] : 0.0f;
        sM[idx] = ok ? max_s[off] : 0.0f;
    }
    __syncthreads();

    for (int o = t; o < 512; o += 256) {
        int oz = o >> 6, oy = (o >> 3) & 7, ox = o & 7;
        float acc = 0.0f;
        for (int dz = 0; dz < 7; ++dz)
            for (int dy = 0; dy < 7; ++dy) {
                int lbase = (oz + dz) * 196 + (oy + dy) * 14 + ox;
                int wbase = dz * 49 + dy * 7;
                #pragma unroll
                for (int dx = 0; dx < 7; ++dx) {
                    acc += sA[lbase + dx] * wl[wbase + dx]
                         + sM[lbase + dx] * wl[343 + wbase + dx];
                }
            }
        int s = ((tz * 8 + oz) * 64 + (ty * 8 + oy)) * 64 + (tx * 8 + ox);
        sig_sa[b * SPAT + s] = sigmoidf_(acc);
    }
}

// ---------------------------------------------------------------------------
// K5: out = x * ca[b][c] * sig_sa[b][s]. Pure float4 streaming.
// ---------------------------------------------------------------------------
__global__ void __launch_bounds__(256) cbam_final(const float4* __restrict__ x4,
                                                  const float* __restrict__ ca,
                                                  const float4* __restrict__ sig4,
                                                  float4* __restrict__ out4) {
    const int gid = blockIdx.x * 256 + threadIdx.x;  // 0 .. 2*64*65536-1
    const int bc = gid >> 16;                        // (b*64+c)
    const int s4 = gid & 65535;
    const int b  = bc >> 6;
    const float g = ca[bc];
    float4 v = x4[gid];
    float4 s = sig4[b * SPAT4 + s4];
    out4[gid] = make_float4(v.x * g * s.x, v.y * g * s.y,
                            v.z * g * s.z, v.w * g * s.w);
}

// ---------------------------------------------------------------------------
// Launch: ws layout (floats):
//   [0,1024)      partial sums   [128][8]
//   [1024,2048)   partial maxes  [128][8]
//   [2048,2176)   ca             [2][64]
//   [4096, +512K) avg_s          [2][64^3/..]  (float4-aligned)
//   next 512K     max_s
//   next 512K     sig_sa
// total ~6.02 MB
// ---------------------------------------------------------------------------
extern "C" void kernel_launch(void* const* d_in, const int* in_sizes, int n_in,
                              void* d_out, int out_size, void* d_ws, size_t ws_size,
                              hipStream_t stream) {
    const float* x   = (const float*)d_in[0];
    const float* w1  = (const float*)d_in[1];
    const float* w2  = (const float*)d_in[2];
    const float* wsp = (const float*)d_in[3];
    float* out = (float*)d_out;
    float* ws  = (float*)d_ws;

    float* p_sum = ws;
    float* p_max = ws + 1024;
    float* p_ca  = ws + 2048;
    float* p_avg = ws + 4096;
    float* p_mx  = ws + 4096 + NB * SPAT;
    float* p_sig = ws + 4096 + 2 * NB * SPAT;

    cbam_stats<<<NB * NC * SPLIT, 256, 0, stream>>>((const float4*)x, p_sum, p_max);
    cbam_mlp_wmma<<<1, 32, 0, stream>>>(p_sum, p_max, w1, w2, p_ca);
    cbam_spatial_stats<<<dim3(SPAT4 / 256, NB), 256, 0, stream>>>(
        (const float4*)x, p_ca, (float4*)p_avg, (float4*)p_mx);
    cbam_conv7<<<dim3(512, NB), 256, 0, stream>>>(p_avg, p_mx, wsp, p_sig);
    cbam_final<<<(NB * NC * SPAT4) / 256, 256, 0, stream>>>(
        (const float4*)x, p_ca, (const float4*)p_sig, (float4*)out);
}